// MatNetCrossMHACaps_25838523253363
// MI455X (gfx1250) — compile-verified
//
#include <hip/hip_runtime.h>
#include <hip/hip_bf16.h>

// MI455X / gfx1250 fused MatNetCrossMHACaps.
// fp32 WMMA (16x16x4) everywhere: workload is HBM/VALU bound (~3.8 GFLOP vs
// ~1.5us bandwidth floor at 23.3 TB/s), so the fp32 matrix path gives full
// reference precision for free. GEMM staging uses CDNA5 async global->LDS
// copies (ASYNCcnt) with double buffering to overlap DMA with WMMA.

typedef __attribute__((ext_vector_type(2))) float v2f;
typedef __attribute__((ext_vector_type(4))) float v4f;
typedef __attribute__((ext_vector_type(8))) float v8f;

#define NB 4      // batch
#define NH 8      // heads
#define NM 512    // q rows
#define NN 512    // kv rows
#define ND 256    // embed dim
#define NDH 32    // head dim

// ---- CDNA5 async global->LDS b128 copy (bypasses VGPRs, tracked by ASYNCcnt)
__device__ __forceinline__ void async_b128_to_lds(const void* gptr, void* sptr)
{
    const unsigned int       l = (unsigned int)(uintptr_t)sptr;       // LDS offset
    const unsigned long long g = (unsigned long long)(uintptr_t)gptr; // global addr
    asm volatile("global_load_async_to_lds_b128 %0, %1, off"
                 :: "v"(l), "v"(g) : "memory");
}
__device__ __forceinline__ void wait_async0()
{
    asm volatile("s_wait_asynccnt 0x0" ::: "memory");
}

// ---------------------------------------------------------------------------
// Generic 64x32-blocktile fp32 WMMA GEMM: C[M,N] = X[M,K] @ W[K,N], K=256.
// 8 waves: 4 (m) x 2 (n) grid of 16x16 WMMA tiles. Async double-buffered LDS.
// mode 0: epilogue scatters to qh[b,h,m,dh]
// mode 1: epilogue splits kv -> kT[b,h,dh,n] (transposed) and vh[b,h,n,dh]
// mode 2: plain row-major nontemporal store
// ---------------------------------------------------------------------------
__global__ __launch_bounds__(256, 2)
void gemm_proj_kernel(const float* __restrict__ X, const float* __restrict__ W,
                      float* __restrict__ out0, float* __restrict__ out1,
                      int K, int N, int mode)
{
    __shared__ float As[2][64 * 16];
    __shared__ float Bs[2][16 * 32];

    const int tid  = threadIdx.x;
    const int lane = tid & 31;
    const int wave = tid >> 5;
    const int wm   = wave & 3;            // m sub-tile 0..3
    const int wn   = wave >> 2;           // n sub-tile 0..1
    const int m0   = blockIdx.x * 64;
    const int n0   = blockIdx.y * 32;

    const int Mloc = lane & 15;           // A row / B col within 16
    const int koff = (lane >> 4) << 1;    // K pair select: 0 or 2
    const int halfsel = (lane >> 4) << 3; // C row offset: 0 or 8

    // cooperative staging maps (one b128 per lane)
    const int arow = tid >> 2, ac4 = (tid & 3) << 2;   // A: 64 rows x 4 float4
    const int brow = tid >> 3, bc4 = (tid & 7) << 2;   // B: 16 rows x 8 float4
    const bool doB = (tid < 128);

    const float* gA = X + (size_t)(m0 + arow) * K + ac4;   // + kk
    const float* gB = W + (size_t)brow * N + n0 + bc4;     // + kk*N
    float* sA[2] = { &As[0][arow * 16 + ac4], &As[1][arow * 16 + ac4] };
    float* sB[2] = { &Bs[0][brow * 32 + bc4], &Bs[1][brow * 32 + bc4] };

    v8f acc = {};

    // prologue: stage K-tile 0 into buffer 0
    async_b128_to_lds(gA, sA[0]);
    if (doB) async_b128_to_lds(gB, sB[0]);
    wait_async0();
    __syncthreads();

    int buf = 0;
    for (int kk = 0; kk < K; kk += 16) {
        // issue next K-tile into the other buffer; DMA overlaps the WMMAs below
        if (kk + 16 < K) {
            async_b128_to_lds(gA + (kk + 16), sA[buf ^ 1]);
            if (doB) async_b128_to_lds(gB + (size_t)(kk + 16) * N, sB[buf ^ 1]);
        }
        const float* Ac = As[buf];
        const float* Bc = Bs[buf];
#pragma unroll
        for (int ks = 0; ks < 16; ks += 4) {
            v2f a, b;
            a.x = Ac[(wm * 16 + Mloc) * 16 + ks + koff];
            a.y = Ac[(wm * 16 + Mloc) * 16 + ks + koff + 1];
            b.x = Bc[(ks + koff) * 32 + wn * 16 + Mloc];
            b.y = Bc[(ks + koff + 1) * 32 + wn * 16 + Mloc];
            acc = __builtin_amdgcn_wmma_f32_16x16x4_f32(
                      false, a, false, b, (short)0, acc, false, false);
        }
        wait_async0();       // next tile's async copies landed
        __syncthreads();     // all waves done reading buf before it is reused
        buf ^= 1;
    }

#pragma unroll
    for (int r = 0; r < 8; ++r) {
        const int g = m0 + wm * 16 + r + halfsel;   // global row
        const int c = n0 + wn * 16 + Mloc;          // global col
        const float val = acc[r];
        if (mode == 2) {
            __builtin_nontemporal_store(val, &out0[(size_t)g * N + c]);
        } else if (mode == 0) {
            const int bi = g >> 9, mr = g & 511;
            const int h = c >> 5, d = c & 31;
            out0[(((size_t)(bi * NH + h)) * NM + mr) * NDH + d] = val;
        } else {
            const int bi = g >> 9, nr = g & 511;
            const int two = c >> 8, rem = c & 255;
            const int h = rem >> 5, d = rem & 31;
            if (two == 0)  // K, stored transposed: kT[b,h,dh,n]
                out0[(((size_t)(bi * NH + h)) * NDH + d) * NN + nr] = val;
            else           // V: vh[b,h,n,dh]
                out1[(((size_t)(bi * NH + h)) * NN + nr) * NDH + d] = val;
        }
    }
}

// ---------------------------------------------------------------------------
// Pointwise 2->16->1 mixer MLP, weights broadcast from LDS.
// w[0..15]=W1[:,0], w[16..31]=W1[:,1], w[32..47]=b1, w[48..63]=W2, w[64]=b2
// ---------------------------------------------------------------------------
__device__ __forceinline__ float mix_eval(float s, float o, const float* __restrict__ w)
{
    float acc = w[64];
#pragma unroll
    for (int k = 0; k < 16; ++k) {
        float hk = fmaf(s, w[k], fmaf(o, w[16 + k], w[32 + k]));
        hk = fmaxf(hk, 0.0f);
        acc = fmaf(hk, w[48 + k], acc);
    }
    return acc;
}

// ---------------------------------------------------------------------------
// Fused attention: one block per (b, h, 16-row m-tile).
// Phase 1: WMMA scores + mixer MLPs -> mixed[16][512] in LDS
// Phase 2: wave-level softmax (shfl_xor reductions, wave32)
// Phase 3: WMMA probs@V, partials reduced with ds_add_f32 LDS atomics
// kT/vh (4 MB total) stay L2-resident across their 32x re-reads; dmat (4 MB,
// re-read by each of the 8 heads) is read nontemporally to avoid evicting them.
// ---------------------------------------------------------------------------
__global__ __launch_bounds__(256, 2)
void attn_mix_kernel(const float* __restrict__ qh, const float* __restrict__ kT,
                     const float* __restrict__ vh,
                     const float* __restrict__ dmat, const float* __restrict__ caps,
                     const float* __restrict__ mixW1, const float* __restrict__ mixb1,
                     const float* __restrict__ mixW2, const float* __restrict__ mixb2,
                     const float* __restrict__ capW1, const float* __restrict__ capb1,
                     const float* __restrict__ capW2, const float* __restrict__ capb2,
                     const float* __restrict__ mix3, const float* __restrict__ mix3bias,
                     float* __restrict__ attn)
{
    __shared__ float sc[16 * 512];    // mixed scores / probs (32 KB)
    __shared__ float qs[16 * 32];     // q tile
    __shared__ float outs[16 * 32];   // PV reduction buffer
    __shared__ float wts[136];        // mixer weights (two MLPs + mix3)

    const int tid  = threadIdx.x;
    const int lane = tid & 31;
    const int wave = tid >> 5;
    const int m0   = blockIdx.x * 16;
    const int h    = blockIdx.y;
    const int bi   = blockIdx.z;

    // stage q tile via CDNA5 async copy (one b128 per lane, 128 lanes)
    if (tid < 128) {
        const int row = tid >> 3, c4 = (tid & 7) << 2;
        async_b128_to_lds(
            qh + (((size_t)(bi * NH + h)) * NM + m0 + row) * NDH + c4,
            &qs[row * 32 + c4]);
    }
    if (tid < 16) {
        wts[tid]       = mixW1[h * 32 + tid];
        wts[16 + tid]  = mixW1[h * 32 + 16 + tid];
        wts[32 + tid]  = mixb1[h * 16 + tid];
        wts[48 + tid]  = mixW2[h * 16 + tid];
        wts[65 + tid]  = capW1[h * 32 + tid];
        wts[81 + tid]  = capW1[h * 32 + 16 + tid];
        wts[97 + tid]  = capb1[h * 16 + tid];
        wts[113 + tid] = capW2[h * 16 + tid];
    } else if (tid == 16) {
        wts[64]  = mixb2[h];
        wts[129] = capb2[h];
        wts[130] = mix3[0];
        wts[131] = mix3[1];
        wts[132] = mix3bias[h];
    }
    outs[tid] = 0.0f;
    outs[tid + 256] = 0.0f;
    wait_async0();
    __syncthreads();

    const int Mloc = lane & 15;
    const int koff = (lane >> 4) << 1;
    const int halfsel = (lane >> 4) << 3;
    const size_t kTbase = ((size_t)(bi * NH + h)) * NDH * NN;
    const size_t vbase  = ((size_t)(bi * NH + h)) * NN * NDH;
    const float scale = 0.17677669529663687f;  // 1/sqrt(32)

    // ---- Phase 1: scores + mixer MLPs; each wave owns a 64-wide n strip ----
#pragma unroll
    for (int t = 0; t < 4; ++t) {
        const int n0 = wave * 64 + t * 16;
        v8f acc = {};
#pragma unroll
        for (int ks = 0; ks < NDH; ks += 4) {
            v2f a, b;
            a.x = qs[Mloc * 32 + ks + koff];
            a.y = qs[Mloc * 32 + ks + koff + 1];
            b.x = kT[kTbase + (size_t)(ks + koff) * NN + n0 + Mloc];
            b.y = kT[kTbase + (size_t)(ks + koff + 1) * NN + n0 + Mloc];
            acc = __builtin_amdgcn_wmma_f32_16x16x4_f32(
                      false, a, false, b, (short)0, acc, false, false);
        }
        const int n = n0 + Mloc;
        const float o2 = caps[bi * NN + n];
#pragma unroll
        for (int r = 0; r < 8; ++r) {
            const int mloc = r + halfsel;
            const int m = m0 + mloc;
            const float s  = acc[r] * scale;
            const float o1 =
                __builtin_nontemporal_load(&dmat[((size_t)(bi * NM + m)) * NN + n]);
            const float dsc = mix_eval(s, o1, &wts[0]);
            const float csc = mix_eval(s, o2, &wts[65]);
            sc[mloc * 512 + n] =
                fmaf(fmaxf(dsc, 0.0f), wts[130],
                     fmaf(fmaxf(csc, 0.0f), wts[131], wts[132]));
        }
    }
    __syncthreads();

    // ---- Phase 2: softmax over n (each wave handles 2 rows, wave32 shfl) ----
#pragma unroll
    for (int rr = 0; rr < 2; ++rr) {
        const int row = wave * 2 + rr;
        float mx = -3.402823e38f;
        for (int n = lane; n < 512; n += 32) mx = fmaxf(mx, sc[row * 512 + n]);
#pragma unroll
        for (int off = 16; off > 0; off >>= 1) mx = fmaxf(mx, __shfl_xor(mx, off, 32));
        float sum = 0.0f;
        for (int n = lane; n < 512; n += 32) {
            const float e = __expf(sc[row * 512 + n] - mx);
            sc[row * 512 + n] = e;
            sum += e;
        }
#pragma unroll
        for (int off = 16; off > 0; off >>= 1) sum += __shfl_xor(sum, off, 32);
        const float rinv = 1.0f / sum;
        for (int n = lane; n < 512; n += 32) sc[row * 512 + n] *= rinv;
    }
    __syncthreads();

    // ---- Phase 3: probs @ V; each wave reduces its 64-wide K strip ----
#pragma unroll
    for (int dt = 0; dt < 2; ++dt) {
        v8f acc = {};
        const int kb = wave * 64;
#pragma unroll
        for (int ks = 0; ks < 64; ks += 4) {
            const int k4 = kb + ks;
            v2f a, b;
            a.x = sc[Mloc * 512 + k4 + koff];
            a.y = sc[Mloc * 512 + k4 + koff + 1];
            b.x = vh[vbase + (size_t)(k4 + koff) * NDH + dt * 16 + Mloc];
            b.y = vh[vbase + (size_t)(k4 + koff + 1) * NDH + dt * 16 + Mloc];
            acc = __builtin_amdgcn_wmma_f32_16x16x4_f32(
                      false, a, false, b, (short)0, acc, false, false);
        }
#pragma unroll
        for (int r = 0; r < 8; ++r)
            atomicAdd(&outs[(r + halfsel) * 32 + dt * 16 + Mloc], acc[r]);
    }
    __syncthreads();

    // store attention output as [b*m, 256] for the final GEMM
    {
        const int i0 = tid, i1 = tid + 256;
        attn[((size_t)(bi * NM + m0 + (i0 >> 5))) * ND + h * NDH + (i0 & 31)] = outs[i0];
        attn[((size_t)(bi * NM + m0 + (i1 >> 5))) * ND + h * NDH + (i1 & 31)] = outs[i1];
    }
}

// ---------------------------------------------------------------------------
extern "C" void kernel_launch(void* const* d_in, const int* in_sizes, int n_in,
                              void* d_out, int out_size, void* d_ws, size_t ws_size,
                              hipStream_t stream)
{
    const float* q_input  = (const float*)d_in[0];
    const float* kv_input = (const float*)d_in[1];
    const float* dmat     = (const float*)d_in[2];
    const float* caps     = (const float*)d_in[3];
    const float* Wq       = (const float*)d_in[4];
    const float* Wkv      = (const float*)d_in[5];
    const float* mixW1    = (const float*)d_in[6];
    const float* mixb1    = (const float*)d_in[7];
    const float* mixW2    = (const float*)d_in[8];
    const float* mixb2    = (const float*)d_in[9];
    const float* capW1    = (const float*)d_in[10];
    const float* capb1    = (const float*)d_in[11];
    const float* capW2    = (const float*)d_in[12];
    const float* capb2    = (const float*)d_in[13];
    const float* mix3     = (const float*)d_in[14];
    const float* mix3bias = (const float*)d_in[15];
    const float* Wout     = (const float*)d_in[16];
    float* out = (float*)d_out;

    const size_t SEG = (size_t)NB * NH * NM * NDH;  // 524288 floats per buffer
    float* qh  = (float*)d_ws;
    float* kTw = qh + SEG;
    float* vhw = kTw + SEG;
    float* atn = vhw + SEG;

    dim3 blk(256);
    // Q projection: [2048,256] @ [256,256] -> qh[b,h,m,dh]
    gemm_proj_kernel<<<dim3(32, 8), blk, 0, stream>>>(
        q_input, Wq, qh, qh, ND, ND, 0);
    // KV projection: [2048,256] @ [256,512] -> kT[b,h,dh,n], vh[b,h,n,dh]
    gemm_proj_kernel<<<dim3(32, 16), blk, 0, stream>>>(
        kv_input, Wkv, kTw, vhw, ND, 2 * ND, 1);
    // Fused scores + mixer MLPs + softmax + PV
    attn_mix_kernel<<<dim3(NM / 16, NH, NB), blk, 0, stream>>>(
        qh, kTw, vhw, dmat, caps,
        mixW1, mixb1, mixW2, mixb2,
        capW1, capb1, capW2, capb2,
        mix3, mix3bias, atn);
    // Output projection: [2048,256] @ [256,256] -> d_out
    gemm_proj_kernel<<<dim3(32, 8), blk, 0, stream>>>(
        atn, Wout, out, out, ND, ND, 2);
}